// DivTree_34797825032802
// MI455X (gfx1250) — compile-verified
//
#include <hip/hip_runtime.h>

// Fused MoE-style 3-stage MLP for MI455X (gfx1250, wave32, WMMA).
// Stage 1: x1 = relu(x0 @ Ws + bs)           (131072x512)x(512x512)
// Stage 2: h  = relu(x1 @ W1[route[a]] + b1) per-agent (4096x512)x(512x512)
// Stage 3: x2 = h @ W2[route[a]] + b2        per-agent (4096x512)x(512x32)
//
// Fusion keeps x1/h tiles in LDS -> ~272 MB HBM traffic instead of ~1.3 GB.
// fp32 accumulation via V_WMMA_F32_16X16X4_F32 preserves reference precision.
// 4x4 register blocking per wave: each B-fragment feeds 4 M-tiles, each
// A-fragment feeds 4 N-tiles; N-columns are partitioned across the 8 waves,
// so each block reads each weight matrix exactly once from the (192 MB) L2.
// Weight prefetch is issued once per 64-deep K-panel at WGP scope.

typedef float v2f __attribute__((ext_vector_type(2)));
typedef float v8f __attribute__((ext_vector_type(8)));

#define B_TOT 4096
#define A_N   32
#define D_IN  512
#define H_DIM 512
#define NA_N  32
#define ROWS  64          // B-rows per block (4 M-tiles of 16)
#define LDSTRIDE 516      // 512 + 4 pad: rows land on distinct LDS banks

// ---- 4 M-tiles x 4 N-tiles register-blocked WMMA over K = 512 ----
// abase: LDS activation buffer (ROWS x LDSTRIDE, row-major), strip 0 base.
// w:     global weight pointer already offset to column n0 (row-major, ld = ldw).
// ISA A layout (16x4 f32): lanes 0-15 hold {K=k,k+1}, lanes 16-31 {K=k+2,k+3}, M = lane&15.
// ISA B layout (4x16 f32): mirrored; N = lane&15, the 2 VGPRs hold consecutive K.
__device__ __forceinline__ void mma_block_4x4(const float* abase, const float* w,
                                              int ldw, v8f acc[4][4])
{
    const int lane = threadIdx.x & 31;
    const int half = lane >> 4;
    const int mrow = lane & 15;
    const float* ap = abase + mrow * LDSTRIDE + (half << 1);
    const float* wp = w + (size_t)(half << 1) * ldw;
#pragma unroll 1
    for (int kp = 0; kp < D_IN; kp += 64) {     // K-panels: one prefetch per panel
        if (kp + 64 < D_IN) {
            // Warm the next K-panel of this wave's weight columns into the WGP
            // cache (near-scope prefetch; source is L2-resident).
            __builtin_prefetch(wp + (size_t)(kp + 64) * ldw, 0, 3);
        }
#pragma unroll 4
        for (int kk = 0; kk < 64; kk += 4) {
            const int k = kp + kk;
            v2f a[4];
#pragma unroll
            for (int m = 0; m < 4; ++m)
                a[m] = *reinterpret_cast<const v2f*>(ap + m * 16 * LDSTRIDE + k);
#pragma unroll
            for (int t = 0; t < 4; ++t) {
                v2f b;
                b.x = wp[(size_t)k * ldw + t * 16];
                b.y = wp[(size_t)k * ldw + ldw + t * 16];
#pragma unroll
                for (int m = 0; m < 4; ++m)
                    acc[m][t] = __builtin_amdgcn_wmma_f32_16x16x4_f32(
                        false, a[m], false, b, (short)0, acc[m][t], false, false);
            }
        }
    }
}

// ---- Single 16x16 tile over K = 512 (used for the narrow GEMM 3) ----
__device__ __forceinline__ v8f mma_tile(const float* abase, const float* w,
                                        int ldw, v8f acc)
{
    const int lane = threadIdx.x & 31;
    const int half = lane >> 4;
    const int mrow = lane & 15;
    const float* ap = abase + mrow * LDSTRIDE + (half << 1);
    const float* wp = w + (size_t)(half << 1) * ldw;
#pragma unroll 8
    for (int k = 0; k < D_IN; k += 4) {
        v2f a = *reinterpret_cast<const v2f*>(ap + k);
        v2f b;
        b.x = wp[(size_t)k * ldw];
        b.y = wp[(size_t)k * ldw + ldw];
        acc = __builtin_amdgcn_wmma_f32_16x16x4_f32(
            false, a, false, b, (short)0, acc, false, false);
    }
    return acc;
}

// Write a 16x16 C/D tile (bias + optional relu) into an LDS strip.
// C/D layout: VGPR i -> row i (lanes 0-15) and row i+8 (lanes 16-31), col = lane&15.
__device__ __forceinline__ void tile_store_lds(float* obase, int n, v8f acc,
                                               const float* __restrict__ bias, bool relu)
{
    const int lane = threadIdx.x & 31;
    const int half = lane >> 4;
    const int col  = lane & 15;
    const float bv = bias[n + col];
#pragma unroll
    for (int i = 0; i < 8; ++i) {
        float v = acc[i] + bv;
        if (relu) v = fmaxf(v, 0.0f);
        obase[(i + half * 8) * LDSTRIDE + n + col] = v;
    }
}

__global__ __launch_bounds__(256)
void divtree_fused_wmma(const float* __restrict__ x0,
                        const float* __restrict__ Ws,
                        const float* __restrict__ bs,
                        const float* __restrict__ W1,
                        const float* __restrict__ b1,
                        const float* __restrict__ W2,
                        const float* __restrict__ b2,
                        const int*   __restrict__ route,
                        float*       __restrict__ out)
{
    extern __shared__ float smem[];
    float* xbuf = smem;                        // ROWS x LDSTRIDE (activations, ping)
    float* hbuf = smem + ROWS * LDSTRIDE;      // ROWS x LDSTRIDE (activations, pong)

    const int a    = blockIdx.y;
    const int row0 = blockIdx.x * ROWS;
    const int tid  = threadIdx.x;
    const int wv   = tid >> 5;                 // wave id, 0..7
    const int r    = route[a];

    // ---- Cooperative load of the x0 tile (ROWS x 512) into LDS (float4) ----
    for (int i = tid; i < ROWS * (D_IN / 4); i += 256) {
        const int rr = i >> 7;                 // / 128
        const int c4 = i & 127;
        const float4 v = *reinterpret_cast<const float4*>(
            x0 + (size_t)(row0 + rr) * (A_N * D_IN) + (size_t)a * D_IN + c4 * 4);
        *reinterpret_cast<float4*>(xbuf + rr * LDSTRIDE + c4 * 4) = v;
    }
    __syncthreads();

    // Each wave owns 4 consecutive N-tiles (n0 = wv*64) and all 4 M-tiles.
    const int n0 = wv * 64;

    // ---- GEMM 1: hbuf = relu(xbuf @ Ws + bs) ----
    {
        v8f acc[4][4] = {};
        mma_block_4x4(xbuf, Ws + n0, H_DIM, acc);
#pragma unroll
        for (int m = 0; m < 4; ++m)
#pragma unroll
            for (int t = 0; t < 4; ++t)
                tile_store_lds(hbuf + m * 16 * LDSTRIDE, n0 + t * 16, acc[m][t], bs, true);
    }
    __syncthreads();

    // ---- GEMM 2: xbuf = relu(hbuf @ W1[r] + b1[r]) ----
    {
        const float* W1g = W1 + (size_t)r * H_DIM * H_DIM;
        const float* b1g = b1 + (size_t)r * H_DIM;
        v8f acc[4][4] = {};
        mma_block_4x4(hbuf, W1g + n0, H_DIM, acc);
#pragma unroll
        for (int m = 0; m < 4; ++m)
#pragma unroll
            for (int t = 0; t < 4; ++t)
                tile_store_lds(xbuf + m * 16 * LDSTRIDE, n0 + t * 16, acc[m][t], b1g, true);
    }
    __syncthreads();

    // ---- GEMM 3: out = xbuf @ W2[r] + b2[r]  (N = 32 -> 4 M x 2 N = 8 tiles, 8 waves) ----
    {
        const int wm = wv & 3;                 // M-tile
        const int wn = wv >> 2;                // N-tile (0..1)
        const float* astrip = xbuf + wm * 16 * LDSTRIDE;
        const float* W2g = W2 + (size_t)r * H_DIM * NA_N;
        const float* b2g = b2 + (size_t)r * NA_N;
        const int n = wn * 16;

        v8f acc = {};
        acc = mma_tile(astrip, W2g + n, NA_N, acc);

        const int lane = tid & 31;
        const int half = lane >> 4;
        const int col  = lane & 15;
        const float bv = b2g[n + col];
#pragma unroll
        for (int i = 0; i < 8; ++i) {
            const int m = wm * 16 + i + half * 8;
            out[((size_t)(row0 + m) * A_N + a) * NA_N + n + col] = acc[i] + bv;
        }
    }
}

extern "C" void kernel_launch(void* const* d_in, const int* in_sizes, int n_in,
                              void* d_out, int out_size, void* d_ws, size_t ws_size,
                              hipStream_t stream) {
    const float* x0    = (const float*)d_in[0];
    const float* Ws    = (const float*)d_in[1];
    const float* bs    = (const float*)d_in[2];
    const float* W1    = (const float*)d_in[3];
    const float* b1    = (const float*)d_in[4];
    const float* W2    = (const float*)d_in[5];
    const float* b2    = (const float*)d_in[6];
    const int*   route = (const int*)d_in[7];
    float* out = (float*)d_out;

    dim3 grid(B_TOT / ROWS, A_N);              // 64 x 32 = 2048 blocks
    dim3 block(256);                           // 8 wave32s
    size_t shmem = (size_t)2 * ROWS * LDSTRIDE * sizeof(float);  // 264 KB < 320 KB WGP LDS

    divtree_fused_wmma<<<grid, block, shmem, stream>>>(x0, Ws, bs, W1, b1, W2, b2, route, out);
}